// SSIMLoss_34746285424741
// MI455X (gfx1250) — compile-verified
//
#include <hip/hip_runtime.h>

typedef float v2f __attribute__((ext_vector_type(2)));
typedef float v8f __attribute__((ext_vector_type(8)));

#define IMG_H 512
#define IMG_W 512
#define BATCH 16
#define WAVES_PER_BLOCK 4
#define TILES_TOTAL (BATCH * (IMG_H / 16) * (IMG_W / 16))   // 16384
#define BLOCKS (TILES_TOTAL / WAVES_PER_BLOCK)              // 4096

// Normalized 11-tap Gaussian, sigma = 1.5 (compile-time constants; match the
// reference's np.exp-based window to float32 precision).
#define W0 0.00102838f
#define W1 0.00759876f
#define W2 0.03600077f
#define W3 0.10936069f
#define W4 0.21300553f
#define W5 0.26601173f

__global__ __launch_bounds__(128) void ssim_tiles(const float* __restrict__ pred,
                                                  const float* __restrict__ targ,
                                                  float* __restrict__ partial) {
    // per-wave LDS slices: 32x28 halo tiles of p and t, 32x20 H bounce buffer
    __shared__ __align__(16) float s_p[WAVES_PER_BLOCK][32][28];
    __shared__ __align__(16) float s_t[WAVES_PER_BLOCK][32][28];
    __shared__ __align__(16) float s_h[WAVES_PER_BLOCK][32][20];
    // banded Gaussian matrix, transposed layout: s_band[n][j] = g[j-n] (0 if out of band)
    __shared__ __align__(16) float s_band[16][28];
    __shared__ float s_red[128];

    const int tid  = threadIdx.x;
    const int lane = tid & 31;
    const int wv   = tid >> 5;
    const int tile = blockIdx.x * WAVES_PER_BLOCK + wv;    // 0 .. 16383
    const int b    = tile >> 10;                           // 1024 tiles per image
    const int rem  = tile & 1023;
    const int oy0  = (rem >> 5) << 4;
    const int ox0  = (rem & 31) << 4;
    const float* __restrict__ pb = pred + (size_t)b * (IMG_H * IMG_W);
    const float* __restrict__ tb = targ + (size_t)b * (IMG_H * IMG_W);

    // ---- build band matrix in LDS (no dynamic register indexing anywhere) ----
    {
        float* bf = &s_band[0][0];
        for (int i = tid; i < 16 * 28; i += 128) bf[i] = 0.0f;
    }
    __syncthreads();
    if (tid < 16) {
        const float wt[11] = {W0, W1, W2, W3, W4, W5, W4, W3, W2, W1, W0};
#pragma unroll
        for (int d = 0; d < 11; ++d)
            s_band[tid][tid + d] = wt[d];   // wt[d] is a compile-time constant
    }

    // ---- stage 0: load 32x28 halo tile (zero-padded, matches 'same' conv) ----
    if (lane < 28) {
        const int gx  = ox0 - 5 + lane;
        const bool cx = (gx >= 0) && (gx < IMG_W);
#pragma unroll 4
        for (int r = 0; r < 32; ++r) {
            const int gy  = oy0 - 5 + r;
            const bool ok = cx && (gy >= 0) && (gy < IMG_H);
            const size_t off = (size_t)gy * IMG_W + gx;
            s_p[wv][r][lane] = ok ? pb[off] : 0.0f;
            s_t[wv][r][lane] = ok ? tb[off] : 0.0f;
        }
    }
    __syncthreads();   // band matrix + tiles visible

    const int nn   = lane & 15;
    const int koff = (lane >= 16) ? 2 : 0;
    const int rhi  = (lane >= 16) ? 8 : 0;

    // band chunks: serve as stage-1 B matrix AND stage-2 A matrix (layouts coincide)
    v2f bnd[7];
#pragma unroll
    for (int c = 0; c < 7; ++c) {
        const int kb = 4 * c + koff;
        bnd[c] = *(const v2f*)&s_band[nn][kb];   // contiguous 8B LDS load
    }

    // ---- five blurred fields: p, t, p*p, t*t, p*t ----
    v8f res[5];
#pragma unroll
    for (int f = 0; f < 5; ++f) {
        // horizontal pass: H[r][n] = sum_j in[r][j] * G[j][n], rows 0..31 via 2 groups
#pragma unroll
        for (int rg = 0; rg < 2; ++rg) {
            v8f acc = {};
#pragma unroll
            for (int c = 0; c < 7; ++c) {
                const int kb  = 4 * c + koff;
                const int row = nn + (rg << 4);
                const v2f pv = *(const v2f*)&s_p[wv][row][kb];
                const v2f tv = *(const v2f*)&s_t[wv][row][kb];
                v2f a;
                if (f == 0)      a = pv;
                else if (f == 1) a = tv;
                else if (f == 2) a = pv * pv;
                else if (f == 3) a = tv * tv;
                else             a = pv * tv;
                acc = __builtin_amdgcn_wmma_f32_16x16x4_f32(
                    false, a, false, bnd[c], (short)0, acc, false, false);
            }
            // C-layout scatter into H bounce buffer
#pragma unroll
            for (int v = 0; v < 8; ++v)
                s_h[wv][v + rhi + (rg << 4)][nn] = acc[v];
        }
        // vertical pass: Out[m][n] = sum_i G[i][m] * H[i][n]
        v8f outv = {};
#pragma unroll
        for (int c = 0; c < 7; ++c) {
            const int kb = 4 * c + koff;
            v2f bb;
            bb[0] = s_h[wv][kb][nn];
            bb[1] = s_h[wv][kb + 1][nn];
            outv = __builtin_amdgcn_wmma_f32_16x16x4_f32(
                false, bnd[c], false, bb, (short)0, outv, false, false);
        }
        res[f] = outv;
    }

    // ---- SSIM map per element (each lane holds 8 output pixels) ----
    float lsum = 0.0f;
#pragma unroll
    for (int e = 0; e < 8; ++e) {
        const float m1  = res[0][e], m2 = res[1][e];
        const float m1s = m1 * m1, m2s = m2 * m2, m12 = m1 * m2;
        const float s1  = res[2][e] - m1s;
        const float s2  = res[3][e] - m2s;
        const float s12 = res[4][e] - m12;
        const float num = (2.0f * m12 + 1.0e-4f) * (2.0f * s12 + 9.0e-4f);
        const float den = (m1s + m2s + 1.0e-4f) * (s1 + s2 + 9.0e-4f);
        lsum += num / den;
    }

    // ---- block reduction -> one partial per block (fixed order: deterministic) ----
    s_red[tid] = lsum;
    __syncthreads();
#pragma unroll
    for (int st = 64; st > 0; st >>= 1) {
        if (tid < st) s_red[tid] += s_red[tid + st];
        __syncthreads();
    }
    if (tid == 0) partial[blockIdx.x] = s_red[0];
}

__global__ void ssim_finalize(const float* __restrict__ partial,
                              float* __restrict__ out, int n) {
    if (threadIdx.x == 0 && blockIdx.x == 0) {
        double s = 0.0;
        for (int i = 0; i < n; ++i) s += (double)partial[i];
        const double npx = (double)BATCH * IMG_H * IMG_W;
        float loss = 1.0f - (float)(s / npx);
        loss = fminf(fmaxf(loss, 0.0f), 2.0f);
        out[0] = loss;
    }
}

extern "C" void kernel_launch(void* const* d_in, const int* in_sizes, int n_in,
                              void* d_out, int out_size, void* d_ws, size_t ws_size,
                              hipStream_t stream) {
    (void)in_sizes; (void)n_in; (void)out_size; (void)ws_size;
    const float* pred = (const float*)d_in[0];
    const float* targ = (const float*)d_in[1];
    float* partial = (float*)d_ws;   // BLOCKS floats of scratch

    ssim_tiles<<<BLOCKS, 128, 0, stream>>>(pred, targ, partial);
    ssim_finalize<<<1, 1, 0, stream>>>(partial, (float*)d_out, BLOCKS);
}